// SPEncoder_33689723470053
// MI455X (gfx1250) — compile-verified
//
#include <hip/hip_runtime.h>

// ---------------------------------------------------------------------------
// Problem constants (from reference)
// ---------------------------------------------------------------------------
static constexpr size_t SL   = 16384;   // sequence length L
static constexpr int    DM   = 192;     // d_model
static constexpr int    DI   = 384;     // d_inner
static constexpr int    DS   = 16;      // d_state
static constexpr int    DCV  = 4;       // d_conv
static constexpr int    NO   = 2;       // num orders
static constexpr int    NB   = 4;       // num blocks (2*NO)
static constexpr float  EPSV = 1e-5f;

// chunked scan config
static constexpr int NCHK = 64;         // chunks
static constexpr int CHL  = 256;        // chunk length (NCHK*CHL == SL)

// ---------------------------------------------------------------------------
// Workspace layout (float offsets). Total ~130M floats (~520 MB).
// ---------------------------------------------------------------------------
static constexpr size_t O_XO    = 0;                                    // NO*SL*DM gathered x
static constexpr size_t O_INV   = O_XO    + (size_t)NO * SL * DM;       // NO*SL    rms inv
static constexpr size_t O_XZ    = O_INV   + (size_t)NO * SL;            // NB*SL*768 in_proj out
static constexpr size_t O_XCV   = O_XZ    + (size_t)NB * SL * 768;      // NB*SL*DI conv out, later y
static constexpr size_t O_DT    = O_XCV   + (size_t)NB * SL * DI;       // NB*SL*DI dt (softplus'd)
static constexpr size_t O_DTRAW = O_DT    + (size_t)NB * SL * DI;       // NB*SL*16 dt_raw (cols 0..11)
static constexpr size_t O_BB    = O_DTRAW + (size_t)NB * SL * 16;       // NB*SL*DS B(t)
static constexpr size_t O_CB    = O_BB    + (size_t)NB * SL * DS;       // NB*SL*DS C(t)
static constexpr size_t O_P     = O_CB    + (size_t)NB * SL * DS;       // NB*NCHK*DS*DI chunk a-prod
static constexpr size_t O_F     = O_P     + (size_t)NB * NCHK * DS * DI;// NB*NCHK*DS*DI chunk local-final
static constexpr size_t O_SI    = O_F     + (size_t)NB * NCHK * DS * DI;// NB*NCHK*DS*DI chunk init state
static constexpr size_t O_OY    = O_SI    + (size_t)NB * NCHK * DS * DI;// NB*SL*DM  y @ Wo
// packed weights: (K/2 pairs, N) -> float2 of rows {2kk, 2kk+1} at col n
static constexpr size_t O_WIP   = O_OY    + (size_t)NB * SL * DM;       // NB*96*768*2
static constexpr size_t O_WXPK  = O_WIP   + (size_t)NB * 96 * 768 * 2;  // NB*192*48*2 (cols 44..47 = 0)
static constexpr size_t O_WDTK  = O_WXPK  + (size_t)NB * 192 * 48 * 2;  // NB*6*384*2
static constexpr size_t O_WOPK  = O_WDTK  + (size_t)NB * 6 * 384 * 2;   // NB*192*192*2
static constexpr size_t O_NWP   = O_WOPK  + (size_t)NB * 192 * 192 * 2; // NB*96*2

// ---------------------------------------------------------------------------
// CDNA5 fp32 WMMA (V_WMMA_F32_16X16X4_F32), wave32.
// A 16x4: lanes 0-15 hold rows m0..m0+15 K={k0,k0+1}; lanes 16-31 K={k0+2,k0+3}
// B 4x16: lane n=l&15; vgpr0=row k0(+2*half), vgpr1=row k0+1(+2*half)
// C/D 16x16: vgpr r -> row m0+r (lanes<16) / m0+r+8 (lanes>=16), col n0+(l&15)
// ---------------------------------------------------------------------------
typedef float v2f __attribute__((ext_vector_type(2)));
typedef float v8f __attribute__((ext_vector_type(8)));

__device__ __forceinline__ v8f wmma4(v2f a, v2f b, v8f c) {
  return __builtin_amdgcn_wmma_f32_16x16x4_f32(
      false, a, false, b, (short)0, c, false, false);
}

__device__ __forceinline__ float wave_sum32(float v) {
#pragma unroll
  for (int m = 16; m >= 1; m >>= 1) v += __shfl_xor(v, m, 32);
  return v;
}

__device__ __forceinline__ float silu(float x) {
  return x / (1.f + __expf(-x));
}

// ---------------------------------------------------------------------------
// pack weights: src (NB, K, srcN) -> dst pairs (NB, K/2, N, 2), zero-pad n>=srcN
// ---------------------------------------------------------------------------
__global__ void k_pack(float* ws, const float* __restrict__ src,
                       size_t dstOff, int K, int N, int srcN) {
  const int idx = blockIdx.x * blockDim.x + threadIdx.x;
  const int total = NB * (K / 2) * N;
  if (idx >= total) return;
  const int n = idx % N;
  const int kk = (idx / N) % (K / 2);
  const int b = idx / (N * (K / 2));
  const float* s = src + (size_t)b * K * srcN;
  const bool ok = (n < srcN);
  ws[dstOff + (size_t)idx * 2 + 0] = ok ? s[(size_t)(2 * kk) * srcN + n] : 0.f;
  ws[dstOff + (size_t)idx * 2 + 1] = ok ? s[(size_t)(2 * kk + 1) * srcN + n] : 0.f;
}

// out = x
__global__ void k_init_out(float* __restrict__ out, const float* __restrict__ x) {
  size_t i = (size_t)blockIdx.x * blockDim.x + threadIdx.x;
  if (i < SL * DM) out[i] = x[i];
}

// ---------------------------------------------------------------------------
// gather + per-row rms scale: xo[o][t] = x[orders[o][t]], invr = rsqrt(mean sq)
// one wave32 per row (6 cols/lane)
// ---------------------------------------------------------------------------
__global__ void k_gather(float* ws, const float* __restrict__ x,
                         const int* __restrict__ orders) {
  const int wave = threadIdx.x >> 5, lane = threadIdx.x & 31;
  const int row = blockIdx.x * 8 + wave;           // 0..NO*SL-1
  const int o = row / (int)SL, t = row % (int)SL;
  const int src = orders[(size_t)o * SL + t];
  const float* xr = x + (size_t)src * DM;
  float v[6], ss = 0.f;
#pragma unroll
  for (int j = 0; j < 6; ++j) { v[j] = xr[lane + j * 32]; ss += v[j] * v[j]; }
  ss = wave_sum32(ss);
  const float inv = rsqrtf(ss / (float)DM + EPSV);
  float* dst = ws + O_XO + ((size_t)o * SL + t) * DM;
#pragma unroll
  for (int j = 0; j < 6; ++j) dst[lane + j * 32] = v[j];
  if (lane == 0) ws[O_INV + (size_t)o * SL + t] = inv;
}

// ---------------------------------------------------------------------------
// GEMM 1: xz[b] = rmsnorm(seq_b) @ Wi[b]   (M=SL, K=192, N=768)
// 32x64 wave tile, 8 fp32-wmma accumulators, packed-B b64 loads.
// grid (64, 12, NB), block 256 (8 waves)
// ---------------------------------------------------------------------------
__global__ void k_gemm_inproj(float* ws) {
  const int b = blockIdx.z;
  const int o = (b < NO) ? b : (b - NO);
  const bool rev = (b >= NO);
  const int wave = threadIdx.x >> 5, lane = threadIdx.x & 31;
  const int half = lane >> 4, l16 = lane & 15;
  const int m0 = (blockIdx.x * 8 + wave) * 32;
  const int n0 = blockIdx.y * 64;
  const int mr0 = m0 + l16, mr1 = m0 + 16 + l16;
  const size_t sr0 = rev ? (SL - 1 - (size_t)mr0) : (size_t)mr0;
  const size_t sr1 = rev ? (SL - 1 - (size_t)mr1) : (size_t)mr1;
  const v2f* arow0 = (const v2f*)(ws + O_XO + ((size_t)o * SL + sr0) * DM);
  const v2f* arow1 = (const v2f*)(ws + O_XO + ((size_t)o * SL + sr1) * DM);
  const float ai0 = ws[O_INV + (size_t)o * SL + sr0];
  const float ai1 = ws[O_INV + (size_t)o * SL + sr1];
  const v2f* nwp = (const v2f*)(ws + O_NWP) + (size_t)b * 96;
  const v2f* Bp = (const v2f*)(ws + O_WIP) + (size_t)b * 96 * 768;
  v8f acc[8] = {};
  for (int k0 = 0; k0 < DM; k0 += 4) {
    const int kk = (k0 >> 1) + half;
    const v2f nw2 = nwp[kk];
    v2f a0 = arow0[kk] * nw2 * ai0;
    v2f a1 = arow1[kk] * nw2 * ai1;
    const v2f* brow = Bp + (size_t)kk * 768 + n0 + l16;
#pragma unroll
    for (int j = 0; j < 4; ++j) {
      const v2f bb = brow[j * 16];
      acc[j] = wmma4(a0, bb, acc[j]);
      acc[4 + j] = wmma4(a1, bb, acc[4 + j]);
    }
  }
  float* dst = ws + O_XZ + (size_t)b * SL * (2 * DI);
#pragma unroll
  for (int j = 0; j < 4; ++j)
#pragma unroll
    for (int r = 0; r < 8; ++r) {
      const int row = m0 + r + 8 * half;
      const int n = n0 + j * 16 + l16;
      dst[(size_t)row * (2 * DI) + n] = acc[j][r];
      dst[(size_t)(row + 16) * (2 * DI) + n] = acc[4 + j][r];
    }
}

// ---------------------------------------------------------------------------
// causal depthwise conv (K=4) + bias + silu : xz[:, :384] -> xcv
// ---------------------------------------------------------------------------
__global__ void k_conv(float* ws, const float* __restrict__ conv_w,
                       const float* __restrict__ conv_b) {
  size_t idx = (size_t)blockIdx.x * blockDim.x + threadIdx.x;
  if (idx >= (size_t)NB * SL * DI) return;
  const int d = idx % DI;
  const size_t t = (idx / DI) % SL;
  const int b = idx / (SL * DI);
  const float* xc = ws + O_XZ + (size_t)b * SL * 768;
  const float* wv = conv_w + ((size_t)b * DI + d) * DCV;
  float acc = conv_b[(size_t)b * DI + d];
#pragma unroll
  for (int k = 0; k < DCV; ++k) {
    long tt = (long)t - 3 + k;
    if (tt >= 0) acc += xc[(size_t)tt * 768 + d] * wv[k];
  }
  ws[O_XCV + (size_t)b * SL * DI + t * DI + d] = silu(acc);
}

// ---------------------------------------------------------------------------
// GEMM 2: proj = xcv @ Wx_pad  (M=SL, K=384, N=48; cols 44..47 zero)
// 32x48 wave tile (6 accums). Epilogue scatters dtraw(0..11)/B(12..27)/C(28..43)
// grid (64, 1, NB)
// ---------------------------------------------------------------------------
__global__ void k_gemm_xproj(float* ws) {
  const int b = blockIdx.z;
  const int wave = threadIdx.x >> 5, lane = threadIdx.x & 31;
  const int half = lane >> 4, l16 = lane & 15;
  const int m0 = (blockIdx.x * 8 + wave) * 32;
  const int mr0 = m0 + l16, mr1 = m0 + 16 + l16;
  const v2f* arow0 = (const v2f*)(ws + O_XCV + ((size_t)b * SL + mr0) * DI);
  const v2f* arow1 = (const v2f*)(ws + O_XCV + ((size_t)b * SL + mr1) * DI);
  const v2f* Bp = (const v2f*)(ws + O_WXPK) + (size_t)b * 192 * 48;
  v8f acc[6] = {};
  for (int k0 = 0; k0 < DI; k0 += 4) {
    const int kk = (k0 >> 1) + half;
    const v2f a0 = arow0[kk];
    const v2f a1 = arow1[kk];
    const v2f* brow = Bp + (size_t)kk * 48 + l16;
#pragma unroll
    for (int j = 0; j < 3; ++j) {
      const v2f bb = brow[j * 16];
      acc[j] = wmma4(a0, bb, acc[j]);
      acc[3 + j] = wmma4(a1, bb, acc[3 + j]);
    }
  }
  float* dtraw = ws + O_DTRAW + (size_t)b * SL * 16;
  float* Bb = ws + O_BB + (size_t)b * SL * DS;
  float* Cb = ws + O_CB + (size_t)b * SL * DS;
#pragma unroll
  for (int j = 0; j < 3; ++j)
#pragma unroll
    for (int r = 0; r < 8; ++r) {
      const int col = j * 16 + l16;
#pragma unroll
      for (int g = 0; g < 2; ++g) {
        const int row = m0 + r + 8 * half + 16 * g;
        const float v = acc[3 * g + j][r];
        if (col < 12)      dtraw[(size_t)row * 16 + col] = v;
        else if (col < 28) Bb[(size_t)row * 16 + (col - 12)] = v;
        else if (col < 44) Cb[(size_t)row * 16 + (col - 28)] = v;
      }
    }
}

// ---------------------------------------------------------------------------
// GEMM 3: dt = softplus(dtraw @ Wdt + bdt)  (M=SL, K=12, N=384)
// 32x64 wave tile, grid (64, 6, NB)
// ---------------------------------------------------------------------------
__global__ void k_gemm_dt(float* ws, const float* __restrict__ dt_b) {
  const int b = blockIdx.z;
  const int wave = threadIdx.x >> 5, lane = threadIdx.x & 31;
  const int half = lane >> 4, l16 = lane & 15;
  const int m0 = (blockIdx.x * 8 + wave) * 32;
  const int n0 = blockIdx.y * 64;
  const v2f* arow0 = (const v2f*)(ws + O_DTRAW + ((size_t)b * SL + m0 + l16) * 16);
  const v2f* arow1 = (const v2f*)(ws + O_DTRAW + ((size_t)b * SL + m0 + 16 + l16) * 16);
  const v2f* Bp = (const v2f*)(ws + O_WDTK) + (size_t)b * 6 * DI;
  v8f acc[8] = {};
  for (int k0 = 0; k0 < 12; k0 += 4) {
    const int kk = (k0 >> 1) + half;
    const v2f a0 = arow0[kk];
    const v2f a1 = arow1[kk];
    const v2f* brow = Bp + (size_t)kk * DI + n0 + l16;
#pragma unroll
    for (int j = 0; j < 4; ++j) {
      const v2f bb = brow[j * 16];
      acc[j] = wmma4(a0, bb, acc[j]);
      acc[4 + j] = wmma4(a1, bb, acc[4 + j]);
    }
  }
  float* dst = ws + O_DT + (size_t)b * SL * DI;
#pragma unroll
  for (int j = 0; j < 4; ++j)
#pragma unroll
    for (int r = 0; r < 8; ++r) {
      const int n = n0 + j * 16 + l16;
      const float bias = dt_b[(size_t)b * DI + n];
#pragma unroll
      for (int g = 0; g < 2; ++g) {
        const int row = m0 + r + 8 * half + 16 * g;
        float v = acc[4 * g + j][r] + bias;
        v = (v > 20.f) ? v : log1pf(__expf(v));
        dst[(size_t)row * DI + n] = v;
      }
    }
}

// ---------------------------------------------------------------------------
// Selective scan, chunked (3 passes). One lane = one channel d, 16 states in
// registers. Per-(d,s) recurrence h = a*h + b with a=exp(dt*A), b=dt*u*B(t).
// ---------------------------------------------------------------------------
__global__ void k_scan1(float* ws, const float* __restrict__ A_log) {
  const int w = blockIdx.x * 4 + (threadIdx.x >> 5);
  const int lane = threadIdx.x & 31;
  const int c = w % NCHK;
  const int g = (w / NCHK) % (DI / 32);
  const int b = w / (NCHK * (DI / 32));
  const int d = g * 32 + lane;
  float As[DS], p[DS], f[DS];
#pragma unroll
  for (int s = 0; s < DS; ++s) {
    As[s] = -__expf(A_log[((size_t)b * DI + d) * DS + s]);
    p[s] = 1.f; f[s] = 0.f;
  }
  const float* dtb = ws + O_DT + (size_t)b * SL * DI;
  const float* ub = ws + O_XCV + (size_t)b * SL * DI;
  const float4* Bb = (const float4*)(ws + O_BB + (size_t)b * SL * DS);
  const int t0 = c * CHL;
  for (int t = t0; t < t0 + CHL; ++t) {
    const float dtv = dtb[(size_t)t * DI + d];
    const float u = ub[(size_t)t * DI + d];
    const float du = dtv * u;
    __builtin_prefetch(&Bb[(t + 8) * 4], 0, 1);
    float4 B0 = Bb[t * 4 + 0], B1 = Bb[t * 4 + 1], B2 = Bb[t * 4 + 2], B3 = Bb[t * 4 + 3];
    const float bs[16] = {B0.x, B0.y, B0.z, B0.w, B1.x, B1.y, B1.z, B1.w,
                          B2.x, B2.y, B2.z, B2.w, B3.x, B3.y, B3.z, B3.w};
#pragma unroll
    for (int s = 0; s < DS; ++s) {
      const float dA = __expf(dtv * As[s]);
      p[s] *= dA;
      f[s] = dA * f[s] + du * bs[s];
    }
  }
#pragma unroll
  for (int s = 0; s < DS; ++s) {
    const size_t i = (((size_t)b * NCHK + c) * DS + s) * DI + d;
    ws[O_P + i] = p[s];
    ws[O_F + i] = f[s];
  }
}

__global__ void k_scan_mid(float* ws) {
  const int g = blockIdx.x * blockDim.x + threadIdx.x;  // over NB*DS*DI
  const int b = g / (DS * DI);
  const int r = g % (DS * DI);
  const int s = r / DI, d = r % DI;
  float S = 0.f;
  for (int c = 0; c < NCHK; ++c) {
    const size_t i = (((size_t)b * NCHK + c) * DS + s) * DI + d;
    ws[O_SI + i] = S;
    S = ws[O_P + i] * S + ws[O_F + i];
  }
}

__global__ void k_scan2(float* ws, const float* __restrict__ A_log,
                        const float* __restrict__ D_p) {
  const int w = blockIdx.x * 4 + (threadIdx.x >> 5);
  const int lane = threadIdx.x & 31;
  const int c = w % NCHK;
  const int g = (w / NCHK) % (DI / 32);
  const int b = w / (NCHK * (DI / 32));
  const int d = g * 32 + lane;
  float As[DS], h[DS];
#pragma unroll
  for (int s = 0; s < DS; ++s) {
    As[s] = -__expf(A_log[((size_t)b * DI + d) * DS + s]);
    h[s] = ws[O_SI + (((size_t)b * NCHK + c) * DS + s) * DI + d];
  }
  const float Dv = D_p[(size_t)b * DI + d];
  const float* dtb = ws + O_DT + (size_t)b * SL * DI;
  float* y = ws + O_XCV + (size_t)b * SL * DI;   // in-place u -> y
  const float* zb = ws + O_XZ + (size_t)b * SL * 768 + DI + d;
  const float4* Bb = (const float4*)(ws + O_BB + (size_t)b * SL * DS);
  const float4* Cb = (const float4*)(ws + O_CB + (size_t)b * SL * DS);
  const int t0 = c * CHL;
  for (int t = t0; t < t0 + CHL; ++t) {
    const float dtv = dtb[(size_t)t * DI + d];
    const float u = y[(size_t)t * DI + d];
    const float du = dtv * u;
    __builtin_prefetch(&Bb[(t + 8) * 4], 0, 1);
    __builtin_prefetch(&Cb[(t + 8) * 4], 0, 1);
    float4 B0 = Bb[t * 4 + 0], B1 = Bb[t * 4 + 1], B2 = Bb[t * 4 + 2], B3 = Bb[t * 4 + 3];
    float4 C0 = Cb[t * 4 + 0], C1 = Cb[t * 4 + 1], C2 = Cb[t * 4 + 2], C3 = Cb[t * 4 + 3];
    const float bs[16] = {B0.x, B0.y, B0.z, B0.w, B1.x, B1.y, B1.z, B1.w,
                          B2.x, B2.y, B2.z, B2.w, B3.x, B3.y, B3.z, B3.w};
    const float cs[16] = {C0.x, C0.y, C0.z, C0.w, C1.x, C1.y, C1.z, C1.w,
                          C2.x, C2.y, C2.z, C2.w, C3.x, C3.y, C3.z, C3.w};
    float yv = 0.f;
#pragma unroll
    for (int s = 0; s < DS; ++s) {
      const float dA = __expf(dtv * As[s]);
      h[s] = dA * h[s] + du * bs[s];
      yv += h[s] * cs[s];
    }
    yv += u * Dv;
    const float zv = zb[(size_t)t * 768];
    yv *= silu(zv);
    y[(size_t)t * DI + d] = yv;
  }
}

// ---------------------------------------------------------------------------
// GEMM 4: oy[b] = y @ Wo[b]   (M=SL, K=384, N=192) 32x64 tile, grid (64,3,NB)
// ---------------------------------------------------------------------------
__global__ void k_gemm_out(float* ws) {
  const int b = blockIdx.z;
  const int wave = threadIdx.x >> 5, lane = threadIdx.x & 31;
  const int half = lane >> 4, l16 = lane & 15;
  const int m0 = (blockIdx.x * 8 + wave) * 32;
  const int n0 = blockIdx.y * 64;
  const v2f* arow0 = (const v2f*)(ws + O_XCV + ((size_t)b * SL + m0 + l16) * DI);
  const v2f* arow1 = (const v2f*)(ws + O_XCV + ((size_t)b * SL + m0 + 16 + l16) * DI);
  const v2f* Bp = (const v2f*)(ws + O_WOPK) + (size_t)b * 192 * DM;
  v8f acc[8] = {};
  for (int k0 = 0; k0 < DI; k0 += 4) {
    const int kk = (k0 >> 1) + half;
    const v2f a0 = arow0[kk];
    const v2f a1 = arow1[kk];
    const v2f* brow = Bp + (size_t)kk * DM + n0 + l16;
#pragma unroll
    for (int j = 0; j < 4; ++j) {
      const v2f bb = brow[j * 16];
      acc[j] = wmma4(a0, bb, acc[j]);
      acc[4 + j] = wmma4(a1, bb, acc[4 + j]);
    }
  }
  float* dst = ws + O_OY + (size_t)b * SL * DM;
#pragma unroll
  for (int j = 0; j < 4; ++j)
#pragma unroll
    for (int r = 0; r < 8; ++r) {
      const int row = m0 + r + 8 * half;
      const int n = n0 + j * 16 + l16;
      dst[(size_t)row * DM + n] = acc[j][r];
      dst[(size_t)(row + 16) * DM + n] = acc[4 + j][r];
    }
}

// ---------------------------------------------------------------------------
// combine per order: f = xo + oy[fb]; b = xo + oy[bb] (time-reversed);
// comb = LN(LN_f(f) + LN_b(b)); out[orders[o][t]] += comb.  One wave per row.
// ---------------------------------------------------------------------------
__global__ void k_combine(float* ws, const int* __restrict__ orders,
                          const float* __restrict__ lwf, const float* __restrict__ lbf,
                          const float* __restrict__ lwb, const float* __restrict__ lbb,
                          const float* __restrict__ lwo, const float* __restrict__ lbo,
                          float* __restrict__ out, int o) {
  const int wave = threadIdx.x >> 5, lane = threadIdx.x & 31;
  const int t = blockIdx.x * 8 + wave;
  const int fb = o, bk = NO + o;
  const float* xr = ws + O_XO + ((size_t)o * SL + t) * DM;
  const float* fy = ws + O_OY + ((size_t)fb * SL + t) * DM;
  const float* by = ws + O_OY + ((size_t)bk * SL + (SL - 1 - (size_t)t)) * DM;
  float fr[6], br[6];
#pragma unroll
  for (int j = 0; j < 6; ++j) {
    const int c = lane + j * 32;
    fr[j] = xr[c] + fy[c];
    br[j] = xr[c] + by[c];
  }
  auto ln = [&](float* v, const float* w, const float* bb) {
    float s = 0.f;
#pragma unroll
    for (int j = 0; j < 6; ++j) s += v[j];
    const float m = wave_sum32(s) / (float)DM;
    float q = 0.f;
#pragma unroll
    for (int j = 0; j < 6; ++j) { const float d0 = v[j] - m; q += d0 * d0; }
    const float rr = rsqrtf(wave_sum32(q) / (float)DM + EPSV);
#pragma unroll
    for (int j = 0; j < 6; ++j) {
      const int c = lane + j * 32;
      v[j] = (v[j] - m) * rr * w[(size_t)o * DM + c] + bb[(size_t)o * DM + c];
    }
  };
  ln(fr, lwf, lbf);
  ln(br, lwb, lbb);
  float cb[6];
#pragma unroll
  for (int j = 0; j < 6; ++j) cb[j] = fr[j] + br[j];
  ln(cb, lwo, lbo);
  const int dst = orders[(size_t)o * SL + t];
#pragma unroll
  for (int j = 0; j < 6; ++j) {
    const int c = lane + j * 32;
    out[(size_t)dst * DM + c] += cb[j];
  }
}

// ---------------------------------------------------------------------------
extern "C" void kernel_launch(void* const* d_in, const int* in_sizes, int n_in,
                              void* d_out, int out_size, void* d_ws, size_t ws_size,
                              hipStream_t stream) {
  (void)in_sizes; (void)n_in; (void)out_size; (void)ws_size;
  const float* x          = (const float*)d_in[0];
  const int*   orders     = (const int*)d_in[1];
  const float* norm_w     = (const float*)d_in[2];
  const float* in_proj_w  = (const float*)d_in[3];
  const float* conv_w     = (const float*)d_in[4];
  const float* conv_b     = (const float*)d_in[5];
  const float* x_proj_w   = (const float*)d_in[6];
  const float* dt_w       = (const float*)d_in[7];
  const float* dt_b       = (const float*)d_in[8];
  const float* A_log      = (const float*)d_in[9];
  const float* D_p        = (const float*)d_in[10];
  const float* out_proj_w = (const float*)d_in[11];
  const float* fwd_ln_w   = (const float*)d_in[12];
  const float* fwd_ln_b   = (const float*)d_in[13];
  const float* bwd_ln_w   = (const float*)d_in[14];
  const float* bwd_ln_b   = (const float*)d_in[15];
  const float* out_ln_w   = (const float*)d_in[16];
  const float* out_ln_b   = (const float*)d_in[17];
  float* out = (float*)d_out;
  float* ws  = (float*)d_ws;

  // pack weights into float2 (k-pair, n) layout for b64 WMMA-B loads
  k_pack<<<dim3((NB * 96 * 768 + 255) / 256), dim3(256), 0, stream>>>(
      ws, in_proj_w, O_WIP, 192, 768, 768);
  k_pack<<<dim3((NB * 192 * 48 + 255) / 256), dim3(256), 0, stream>>>(
      ws, x_proj_w, O_WXPK, 384, 48, 44);
  k_pack<<<dim3((NB * 6 * 384 + 255) / 256), dim3(256), 0, stream>>>(
      ws, dt_w, O_WDTK, 12, 384, 384);
  k_pack<<<dim3((NB * 192 * 192 + 255) / 256), dim3(256), 0, stream>>>(
      ws, out_proj_w, O_WOPK, 384, 192, 192);
  k_pack<<<dim3((NB * 96 + 255) / 256), dim3(256), 0, stream>>>(
      ws, norm_w, O_NWP, 192, 1, 1);

  k_init_out<<<dim3((unsigned)((SL * DM + 255) / 256)), dim3(256), 0, stream>>>(out, x);
  k_gather<<<dim3((unsigned)(NO * SL / 8)), dim3(256), 0, stream>>>(ws, x, orders);

  k_gemm_inproj<<<dim3(64, 12, NB), dim3(256), 0, stream>>>(ws);
  k_conv<<<dim3((unsigned)((NB * SL * DI + 255) / 256)), dim3(256), 0, stream>>>(ws, conv_w, conv_b);
  k_gemm_xproj<<<dim3(64, 1, NB), dim3(256), 0, stream>>>(ws);
  k_gemm_dt<<<dim3(64, 6, NB), dim3(256), 0, stream>>>(ws, dt_b);

  // chunked selective scan (3 passes)
  k_scan1<<<dim3(NB * (DI / 32) * NCHK / 4), dim3(128), 0, stream>>>(ws, A_log);
  k_scan_mid<<<dim3(NB * DS * DI / 256), dim3(256), 0, stream>>>(ws);
  k_scan2<<<dim3(NB * (DI / 32) * NCHK / 4), dim3(128), 0, stream>>>(ws, A_log, D_p);

  k_gemm_out<<<dim3(64, 3, NB), dim3(256), 0, stream>>>(ws);

  for (int o = 0; o < NO; ++o)
    k_combine<<<dim3((unsigned)(SL / 8)), dim3(256), 0, stream>>>(
        ws, orders, fwd_ln_w, fwd_ln_b, bwd_ln_w, bwd_ln_b, out_ln_w, out_ln_b, out, o);
}